// ComEmb_84851373900030
// MI455X (gfx1250) — compile-verified
//
#include <hip/hip_runtime.h>
#include <hip/hip_bf16.h>

typedef __attribute__((ext_vector_type(2))) float v2f;
typedef __attribute__((ext_vector_type(8))) float v8f;

#define CCH 128  // channels

// Native f32 atomic add (single global_atomic_add_f32, no CAS loop).
__device__ __forceinline__ void atomic_add_f32(float* p, float v) {
#if defined(__HIP_DEVICE_COMPILE__)
    unsafeAtomicAdd(p, v);
#else
    atomicAdd(p, v);
#endif
}

// ---------------------------------------------------------------------------
// 1) init: zero accumulator (d_out) over N*C, and deg[i] = 1.0 (self loop)
// ---------------------------------------------------------------------------
__global__ void init_kernel(float* __restrict__ acc, float* __restrict__ deg,
                            int n, int nc) {
    int i = blockIdx.x * blockDim.x + threadIdx.x;
    if (i < nc) acc[i] = 0.0f;
    if (i < n)  deg[i] = 1.0f;   // self-loop weight
}

// ---------------------------------------------------------------------------
// 2) degree accumulation: deg[col[e]] += ew[e]
// ---------------------------------------------------------------------------
__global__ void degree_kernel(const int* __restrict__ col,
                              const float* __restrict__ ew,
                              float* __restrict__ deg, int e_cnt) {
    int e = blockIdx.x * blockDim.x + threadIdx.x;
    if (e < e_cnt) atomic_add_f32(&deg[col[e]], ew[e]);
}

// ---------------------------------------------------------------------------
// 3) dinv = deg > 0 ? rsqrt(deg) : 0   (in place)
// ---------------------------------------------------------------------------
__global__ void dinv_kernel(float* __restrict__ deg, int n) {
    int i = blockIdx.x * blockDim.x + threadIdx.x;
    if (i < n) {
        float d = deg[i];
        deg[i] = (d > 0.0f) ? __frsqrt_rn(d) : 0.0f;
    }
}

// ---------------------------------------------------------------------------
// 4) h = embs @ W via V_WMMA_F32_16X16X4_F32  (exact fp32, matches reference)
//    Block = 256 threads = 8 waves. Each block: 16 rows x 128 cols.
//    Wave w computes the 16x16 tile at columns [16w, 16w+16).
//    A (16x4 f32, 2 VGPR):  lane<16 -> K=k,k+1 ; lane>=16 -> K=k+2,k+3, M=lane%16
//    B (4x16 f32, 2 VGPR):  same K split, N=lane%16
//    C/D (16x16 f32, 8 VGPR): row M = 8*(lane/16)+v, col N = lane%16
// ---------------------------------------------------------------------------
__global__ void gemm_wmma_kernel(const float* __restrict__ embs,
                                 const float* __restrict__ W,
                                 float* __restrict__ h, int n) {
    const int wave = threadIdx.x >> 5;        // 0..7 -> column tile
    const int lane = threadIdx.x & 31;
    const int row0 = blockIdx.x * 16;
    const int col0 = wave * 16;
    const int m    = lane & 15;
    const int koff = (lane >> 4) * 2;         // 0 or 2

    int arow = row0 + m;
    if (arow >= n) arow = n - 1;              // clamp (n % 16 == 0 here anyway)
    const float* __restrict__ ap = embs + (size_t)arow * CCH;
    const float* __restrict__ bp = W + (col0 + m);

    v8f c = {};
    #pragma unroll
    for (int k = 0; k < CCH; k += 4) {
        v2f a, b;
        a.x = ap[k + koff];
        a.y = ap[k + koff + 1];
        b.x = bp[(size_t)(k + koff)     * CCH];
        b.y = bp[(size_t)(k + koff + 1) * CCH];
        c = __builtin_amdgcn_wmma_f32_16x16x4_f32(
                /*neg_a=*/false, a, /*neg_b=*/false, b,
                /*c_mod=*/(short)0, c, /*reuse_a=*/false, /*reuse_b=*/false);
    }

    const int mbase = (lane >> 4) * 8;
    const int ncol  = lane & 15;
    #pragma unroll
    for (int i = 0; i < 8; ++i) {
        int r = row0 + mbase + i;
        if (r < n) h[(size_t)r * CCH + col0 + ncol] = c[i];
    }
}

// ---------------------------------------------------------------------------
// 5) edge scatter: one wave32 per edge. Each lane handles 4 channels:
//    b128 gather of h[row], scale by norm, 4x global_atomic_add_f32 to acc[col]
// ---------------------------------------------------------------------------
__global__ void scatter_kernel(const float* __restrict__ h,
                               const int* __restrict__ row,
                               const int* __restrict__ col,
                               const float* __restrict__ ew,
                               const float* __restrict__ dinv,
                               float* __restrict__ acc, int e_cnt) {
    const int e = blockIdx.x * 8 + (threadIdx.x >> 5);   // 8 waves / block
    if (e >= e_cnt) return;
    const int lane = threadIdx.x & 31;

    const int r = row[e];
    const int t = col[e];
    const float nrm = dinv[r] * ew[e] * dinv[t];

    const float4 v = ((const float4*)(h + (size_t)r * CCH))[lane];
    float* ap = acc + (size_t)t * CCH + lane * 4;
    atomic_add_f32(ap + 0, nrm * v.x);
    atomic_add_f32(ap + 1, nrm * v.y);
    atomic_add_f32(ap + 2, nrm * v.z);
    atomic_add_f32(ap + 3, nrm * v.w);
}

// ---------------------------------------------------------------------------
// 6) final fuse: out = (acc + dinv^2 * h) * X, float4-vectorized.
//    Self-loop message (norm = dinv*1*dinv) folded in -> no atomics needed.
// ---------------------------------------------------------------------------
__global__ void final_kernel(float* __restrict__ out,       // acc in, result out
                             const float* __restrict__ h,
                             const float* __restrict__ dinv,
                             const float* __restrict__ X, int nq) {
    const int i = blockIdx.x * blockDim.x + threadIdx.x;   // over N*C/4
    if (i >= nq) return;
    const int node = i >> 5;                               // (i*4)/128
    const float di = dinv[node];
    const float s = di * di;
    float4 a = ((float4*)out)[i];
    const float4 hv = ((const float4*)h)[i];
    const float4 xv = ((const float4*)X)[i];
    a.x = (a.x + s * hv.x) * xv.x;
    a.y = (a.y + s * hv.y) * xv.y;
    a.z = (a.z + s * hv.z) * xv.z;
    a.w = (a.w + s * hv.w) * xv.w;
    ((float4*)out)[i] = a;
}

extern "C" void kernel_launch(void* const* d_in, const int* in_sizes, int n_in,
                              void* d_out, int out_size, void* d_ws, size_t ws_size,
                              hipStream_t stream) {
    const float* X    = (const float*)d_in[0];
    const float* embs = (const float*)d_in[1];
    const float* W    = (const float*)d_in[2];
    const int*   ei   = (const int*)d_in[3];
    const float* ew   = (const float*)d_in[4];

    const int nc    = in_sizes[0];          // N*C
    const int n     = nc / CCH;             // N
    const int e_cnt = in_sizes[4];          // E
    const int* row  = ei;                   // edge_index[0]
    const int* col  = ei + e_cnt;           // edge_index[1]

    float* out = (float*)d_out;             // doubles as scatter accumulator
    float* h   = (float*)d_ws;              // N*C floats
    float* deg = h + (size_t)nc;            // N floats (becomes dinv in place)

    // 1) zero acc, deg = 1.0 (self loop)
    init_kernel<<<(nc + 255) / 256, 256, 0, stream>>>(out, deg, n, nc);
    // 2) degree
    degree_kernel<<<(e_cnt + 255) / 256, 256, 0, stream>>>(col, ew, deg, e_cnt);
    // 3) dinv in place
    dinv_kernel<<<(n + 255) / 256, 256, 0, stream>>>(deg, n);
    // 4) h = embs @ W  (WMMA fp32)
    gemm_wmma_kernel<<<(n + 15) / 16, 256, 0, stream>>>(embs, W, h, n);
    // 5) edge scatter (L2-resident h/acc, native f32 atomics)
    scatter_kernel<<<(e_cnt + 7) / 8, 256, 0, stream>>>(h, row, col, ew, deg, out, e_cnt);
    // 6) fused self-loop + gating
    final_kernel<<<(nc / 4 + 255) / 256, 256, 0, stream>>>(out, h, deg, X, nc / 4);
}